// FVRF_25632364823088
// MI455X (gfx1250) — compile-verified
//
#include <hip/hip_runtime.h>

typedef float v2f __attribute__((ext_vector_type(2)));
typedef float v8f __attribute__((ext_vector_type(8)));

#define R    128
#define NV   2000
#define KM1  30
#define KC   31          // K = KM1 + 1
#define RK   (R * KM1)   // 3840

// ---------------------------------------------------------------------------
// Kernel 1: Post_mean_coefs = [C_mu | X @ W], plus cov-matrix border init
// (d_out is poisoned by the harness, so [v,0,0]=0.5 and the zero row/col of
// each 31x31 cov matrix must be written on every call).
// ---------------------------------------------------------------------------
__global__ void mean_init_kernel(const float* __restrict__ Xi,   // (R, NV)
                                 const float* __restrict__ Wm,   // (R*KM1, 1)
                                 const float* __restrict__ Cmu,  // (NV, 1)
                                 float* __restrict__ out_mean,   // (NV, KC)
                                 float* __restrict__ out_cov) {  // (NV, KC, KC)
    int idx = blockIdx.x * blockDim.x + threadIdx.x;
    if (idx >= NV * KC) return;
    int v = idx / KC;
    int k = idx % KC;
    size_t covb = (size_t)v * KC * KC;
    if (k == 0) {
        out_mean[(size_t)v * KC] = Cmu[v];
        out_cov[covb] = 0.5f;             // SIGMA2_MU
    } else {
        int km = k - 1;
        float s = 0.f;
        #pragma unroll 8
        for (int i = 0; i < R; ++i)
            s += Xi[(size_t)i * NV + v] * Wm[i * KM1 + km];
        out_mean[(size_t)v * KC + k] = s;
        out_cov[covb + k]      = 0.0f;    // row 0
        out_cov[covb + k * KC] = 0.0f;    // col 0
    }
}

// ---------------------------------------------------------------------------
// Kernel 2: Post_cov_c[v,k,l] = sum_ij X[v,i] * cov[i*30+k, j*30+l] * X[v,j]
// One block per (k,l) pair (x 4 v-splits). B_{kl} (128x128) staged in LDS,
// pair-interleaved so each lane's 2-VGPR B fragment is one 8-byte DS read
// (compiler fuses fragment pairs into ds_load_2addr_b64).
// Each wave computes two 16x16 Y tiles concurrently with independent
// accumulator chains (c0/c1) sharing the register-resident A fragments
// (2x WMMA ILP, no spills at ~115 VGPRs). Residual LDS-fetch latency between
// WMMA pairs is hidden by wave-level parallelism: 512-thread blocks put
// 4+ waves per SIMD, i.e. >=8 independent WMMA chains in flight per SIMD.
// The second contraction (rowsum of Y .* X) is fused via the documented
// C/D VGPR layout + wave32 cross-lane shuffles.
// ---------------------------------------------------------------------------
__global__ void cov_kernel(const float* __restrict__ Xi,      // (R, NV)
                           const float* __restrict__ cov,     // (RK, RK)
                           float* __restrict__ out_cov) {     // (NV, KC, KC)
    extern __shared__ float ldsB[];      // 128*128 floats, pair-interleaved
    const int kk = blockIdx.x / KM1;     // k index (0..29)
    const int ll = blockIdx.x % KM1;     // l index (0..29)
    const int tid = threadIdx.x;

    // Gather B_{kl}[i][j] = cov[(i*30+kk)*3840 + j*30+ll] into LDS.
    // Layout: ldsB[ ((i>>1)*128 + j)*2 + (i&1) ]  -> (B[2q][j], B[2q+1][j])
    // adjacent, so a B fragment row-pair is a single 8-byte LDS read.
    for (int e = tid; e < R * R; e += blockDim.x) {
        int i = e >> 7;
        int j = e & 127;
        float val = cov[(size_t)(i * KM1 + kk) * RK + (size_t)j * KM1 + ll];
        ldsB[(((i >> 1) * R + j) << 1) | (i & 1)] = val;
    }
    __syncthreads();

    const int lane  = tid & 31;
    const int wave  = tid >> 5;                 // 0..15
    const int m     = lane & 15;                // column-within-16 / row id
    const int hi    = lane >> 4;                // 0 or 1 (lane half)
    const int gwave = blockIdx.y * 16 + wave;   // 0..63 global wave slot
    const v2f* ldsB2 = (const v2f*)ldsB;

    for (int vt = gwave; vt < NV / 16; vt += 64) {
        const int v0 = vt * 16;

        // A fragments for the whole K=128 (32 steps of K=4), kept in VGPRs.
        // ISA 32-bit A 16x4 layout: VGPR0 = K {0 | 2}, VGPR1 = K {1 | 3}
        // across lane halves; M = lane&15.  X[v,i] = Xi[i*NV + v].
        v2f a[32];
        #pragma unroll
        for (int t = 0; t < 32; ++t) {
            const int krow = 4 * t + 2 * hi;
            const float* p = Xi + (size_t)krow * NV + v0 + m;
            a[t][0] = p[0];     // K = krow
            a[t][1] = p[NV];    // K = krow + 1
        }

        float part[8];
        #pragma unroll
        for (int p = 0; p < 8; ++p) part[p] = 0.f;

        // Loop over the 8 N-tiles of Y = X_tile @ B, two tiles at a time so
        // the two WMMA accumulation chains are independent (2x ILP).
        for (int j0 = 0; j0 < 8; j0 += 2) {
            v8f c0 = {0.f, 0.f, 0.f, 0.f, 0.f, 0.f, 0.f, 0.f};
            v8f c1 = {0.f, 0.f, 0.f, 0.f, 0.f, 0.f, 0.f, 0.f};
            #pragma unroll
            for (int t = 0; t < 32; ++t) {
                // B fragments: rows (4t+2*hi, 4t+2*hi+1); cols of two tiles,
                // 128 bytes apart in LDS (one ds_load_2addr_b64).
                const int row = (2 * t + hi) * R + m;
                v2f b0 = ldsB2[row + j0 * 16];
                v2f b1 = ldsB2[row + j0 * 16 + 16];
                c0 = __builtin_amdgcn_wmma_f32_16x16x4_f32(
                        false, a[t], false, b0, (short)0, c0, false, false);
                c1 = __builtin_amdgcn_wmma_f32_16x16x4_f32(
                        false, a[t], false, b1, (short)0, c1, false, false);
            }
            // Fused second contraction: part[p] += Y[M,N] * X[v0+M, j*16+N]
            // C/D layout: VGPR p -> M = p + 8*hi, N = m.
            #pragma unroll
            for (int p = 0; p < 8; ++p) {
                float xe0 = Xi[(size_t)(j0 * 16 + m) * NV + v0 + p + 8 * hi];
                float xe1 = Xi[(size_t)(j0 * 16 + 16 + m) * NV + v0 + p + 8 * hi];
                part[p] += c0[p] * xe0 + c1[p] * xe1;
            }
        }

        // Reduce over N: sum across the 16 lanes of each half (wave32 shuffles).
        #pragma unroll
        for (int p = 0; p < 8; ++p) {
            float s = part[p];
            s += __shfl_xor(s, 1, 16);
            s += __shfl_xor(s, 2, 16);
            s += __shfl_xor(s, 4, 16);
            s += __shfl_xor(s, 8, 16);
            part[p] = s;
        }

        // Lane 0 writes rows v0..v0+7, lane 16 writes v0+8..v0+15.
        if (m == 0) {
            #pragma unroll
            for (int p = 0; p < 8; ++p) {
                int v = v0 + p + 8 * hi;
                out_cov[(size_t)v * KC * KC + (size_t)(1 + kk) * KC + (1 + ll)]
                    = part[p];
            }
        }
    }
}

// ---------------------------------------------------------------------------
extern "C" void kernel_launch(void* const* d_in, const int* in_sizes, int n_in,
                              void* d_out, int out_size, void* d_ws, size_t ws_size,
                              hipStream_t stream) {
    const float* Xi  = (const float*)d_in[0];   // Xi_evals (128, 2000)
    const float* Wm  = (const float*)d_in[1];   // vec_W_post_mean (3840, 1)
    const float* cov = (const float*)d_in[2];   // vec_W_post_cov (3840, 3840)
    const float* Cmu = (const float*)d_in[3];   // C_mu (2000, 1)

    float* out_mean = (float*)d_out;              // (2000, 31)
    float* out_cov  = out_mean + (size_t)NV * KC; // (2000, 31, 31)

    // Mean coefs + cov border init (independent of cov_kernel's region).
    {
        int n = NV * KC;
        mean_init_kernel<<<(n + 255) / 256, 256, 0, stream>>>(
            Xi, Wm, Cmu, out_mean, out_cov);
    }

    // 900 (k,l) pairs x 4 v-splits; 512-thread blocks (16 waves) for
    // wave-level latency hiding; 64 KB dynamic LDS for the B_{kl} tile.
    cov_kernel<<<dim3(KM1 * KM1, 4), 512, R * R * sizeof(float), stream>>>(
        Xi, cov, out_cov);
}